// yoloLoss_44315472560524
// MI455X (gfx1250) — compile-verified
//
#include <hip/hip_runtime.h>
#include <hip/hip_bf16.h>

// ---------------------------------------------------------------------------
// YOLO-v1 loss, MI455X (gfx1250, wave32).
// Memory-bound streaming reduction: 96 MB read -> 1 scalar. One thread per
// (n,i,j) cell, float2-vectorized loads, then a deterministic reduction tree
// whose wave-level step uses V_WMMA_F32_16X16X4_F32 with an all-ones B matrix
// (exact f32 row-sum reduce: 32 lane values -> 1 in a single matrix op).
// ---------------------------------------------------------------------------

typedef __attribute__((ext_vector_type(2))) float v2f;
typedef __attribute__((ext_vector_type(8))) float v8f;

#define S_GRID 7
#define CH     30
#define L_COORD 5.0f
#define L_NOOBJ 0.5f
#define CELL_PX 64.0f
#define IMG_PX  448.0f

// Full-wave reduction of one float per lane using the f32 WMMA unit.
// A[16x4] holds the 32 lane values (a.x = val, a.y = 0); B[4x16] = ones.
// D row m = val_m + val_{m+16}. D VGPR r: lanes 0-15 hold M=r, lanes 16-31
// hold M=8+r, so summing the 8 D registers gives half-sums per lane group;
// one xor-16 shuffle completes the reduce. Exact f32 (FMA accumulation).
// Requires EXEC == all ones (call from wave-uniform control flow only).
__device__ __forceinline__ float wave_reduce_wmma(float val) {
    v2f a; a.x = val;  a.y = 0.0f;
    v2f b; b.x = 1.0f; b.y = 1.0f;
    v8f c = {};
    v8f d = __builtin_amdgcn_wmma_f32_16x16x4_f32(
        /*neg_a=*/false, a, /*neg_b=*/false, b,
        /*c_mod=*/(short)0, c, /*reuse_a=*/false, /*reuse_b=*/false);
    float s = ((d[0] + d[1]) + (d[2] + d[3])) + ((d[4] + d[5]) + (d[6] + d[7]));
    s += __shfl_xor(s, 16, 32);   // combine lanes 0-15 half with 16-31 half
    return s;                     // every lane now holds the full 32-lane sum
}

// 256-thread (8-wave) block reduce; result valid in wave 0 (all its lanes).
__device__ __forceinline__ float block_reduce_wmma(float val) {
    __shared__ float lds[8];
    const int lane = threadIdx.x & 31;
    const int wid  = threadIdx.x >> 5;
    float w = wave_reduce_wmma(val);
    if (lane == 0) lds[wid] = w;
    __syncthreads();
    float r = 0.0f;
    if (wid == 0) {                                  // wave-uniform branch
        float v = (lane < 8) ? lds[lane] : 0.0f;
        r = wave_reduce_wmma(v);
    }
    return r;
}

__global__ void __launch_bounds__(256)
yolo_loss_cells(const float* __restrict__ pred,
                const float* __restrict__ targ,
                float* __restrict__ partials,
                int num_cells) {
    const int cell = blockIdx.x * 256 + threadIdx.x;

    float cell_loss = 0.0f;
    if (cell < num_cells) {
        // cell byte offset = cell*120 -> always 8-byte aligned: float2 loads.
        const float2* p2 = reinterpret_cast<const float2*>(pred) + (size_t)cell * (CH / 2);
        const float2* t2 = reinterpret_cast<const float2*>(targ) + (size_t)cell * (CH / 2);
        float pr[CH], tg[CH];
#pragma unroll
        for (int k = 0; k < CH / 2; ++k) {
            float2 v = p2[k]; pr[2 * k] = v.x; pr[2 * k + 1] = v.y;
            float2 u = t2[k]; tg[2 * k] = u.x; tg[2 * k + 1] = u.y;
        }

        const float coo = (tg[4] > 0.0f) ? 1.0f : 0.0f;
        const float noo = (tg[4] == 0.0f) ? 1.0f : 0.0f;

        // ---- no-object loss: conf channels 4 and 9 of both pred boxes
        const float d4 = pr[4] - tg[4];
        const float d9 = pr[9] - tg[9];
        const float nooobj = noo * (d4 * d4 + d9 * d9);

        // ---- target box 0 in xyxy
        const float tcx = tg[0] * CELL_PX, tcy = tg[1] * CELL_PX;
        const float thw = 0.5f * tg[2] * IMG_PX, thh = 0.5f * tg[3] * IMG_PX;
        const float tx1 = tcx - thw, ty1 = tcy - thh;
        const float tx2 = tcx + thw, ty2 = tcy + thh;
        const float area_t = (tx2 - tx1) * (ty2 - ty1);

        // ---- IoU of each pred box vs target box 0
        float iou[2];
#pragma unroll
        for (int bx = 0; bx < 2; ++bx) {
            const int off = bx * 5;
            const float cx = pr[off + 0] * CELL_PX, cy = pr[off + 1] * CELL_PX;
            const float hw = 0.5f * pr[off + 2] * IMG_PX, hh = 0.5f * pr[off + 3] * IMG_PX;
            const float x1 = cx - hw, y1 = cy - hh, x2 = cx + hw, y2 = cy + hh;
            const float iw = fminf(x2, tx2) - fmaxf(x1, tx1);
            const float ih = fminf(y2, ty2) - fmaxf(y1, ty1);
            const float inter = fmaxf(iw, 0.0f) * fmaxf(ih, 0.0f);
            const float area_p = (x2 - x1) * (y2 - y1);
            iou[bx] = inter / (area_p + area_t - inter);
        }

        // argmax (first index wins ties, matching jnp.argmax)
        const int   idx     = (iou[1] > iou[0]) ? 1 : 0;
        const float max_iou = fmaxf(iou[0], iou[1]);
        const int   ro  = idx * 5;        // responsible pred/target box
        const int   nro = (1 - idx) * 5;  // non-responsible pred box

        const float dc = pr[ro + 4] - max_iou;
        const float contain = coo * dc * dc;

        const float dx = pr[ro + 0] - tg[ro + 0];
        const float dy = pr[ro + 1] - tg[ro + 1];
        const float dw = sqrtf(pr[ro + 2]) - sqrtf(tg[ro + 2]);
        const float dh = sqrtf(pr[ro + 3]) - sqrtf(tg[ro + 3]);
        const float loc = coo * (dx * dx + dy * dy + dw * dw + dh * dh);

        const float not_contain = coo * pr[nro + 4] * pr[nro + 4];

        float cls = 0.0f;
#pragma unroll
        for (int cch = 10; cch < CH; ++cch) {
            const float dd = pr[cch] - tg[cch];
            cls += dd * dd;
        }
        cls *= coo;

        cell_loss = L_COORD * loc + contain + not_contain + L_NOOBJ * nooobj + cls;
    }

    // reconverged here: EXEC all-ones for the WMMA reduce
    const float bsum = block_reduce_wmma(cell_loss);
    if (threadIdx.x == 0) partials[blockIdx.x] = bsum;
}

__global__ void __launch_bounds__(256)
yolo_loss_final(const float* __restrict__ partials,
                float* __restrict__ out,
                int num_partials, float inv_n) {
    float acc = 0.0f;
    for (int i = threadIdx.x; i < num_partials; i += 256)
        acc += partials[i];
    const float total = block_reduce_wmma(acc);
    if (threadIdx.x == 0) out[0] = total * inv_n;
}

extern "C" void kernel_launch(void* const* d_in, const int* in_sizes, int n_in,
                              void* d_out, int out_size, void* d_ws, size_t ws_size,
                              hipStream_t stream) {
    const float* pred = (const float*)d_in[0];
    const float* targ = (const float*)d_in[1];
    float* out      = (float*)d_out;
    float* partials = (float*)d_ws;

    const int n_batch   = in_sizes[0] / (S_GRID * S_GRID * CH);  // 8192
    const int num_cells = n_batch * S_GRID * S_GRID;             // 401408
    const int blocks    = (num_cells + 255) / 256;               // 1568
    const float inv_n   = 1.0f / (float)n_batch;

    yolo_loss_cells<<<blocks, 256, 0, stream>>>(pred, targ, partials, num_cells);
    yolo_loss_final<<<1, 256, 0, stream>>>(partials, out, blocks, inv_n);
}